// STGMNBlock_58050777972949
// MI455X (gfx1250) — compile-verified
//
#include <hip/hip_runtime.h>
#include <math.h>

// ---------------------------------------------------------------------------
// STGMN block forward for MI455X (gfx1250, wave32).
// * All GEMMs through V_WMMA_F32_16X16X4_F32 (fp32, matches reference).
// * Hot GEMM (E = e2 @ S, 64 x 512^3) uses double-buffered LDS tiles fed by
//   GLOBAL_LOAD_ASYNC_TO_LDS_B128 (ASYNCcnt / s_wait_asynccnt) so data motion
//   bypasses VGPRs and overlaps WMMA issue.
// * No atomics anywhere -> deterministic, graph-capture safe.
// ---------------------------------------------------------------------------

typedef __attribute__((ext_vector_type(2))) float v2f;
typedef __attribute__((ext_vector_type(8))) float v8f;

#define BB 64
#define NN 512
#define TT 16
#define DD 64
#define TO 12
#define ROWS_PER_B (NN * TO)          // 6144 rows of [64] per batch

__device__ __forceinline__ v8f wmma4(v2f a, v2f b, v8f c) {
  // D = A(16x4) x B(4x16) + C(16x16), fp32
  return __builtin_amdgcn_wmma_f32_16x16x4_f32(false, a, false, b, (short)0, c,
                                               false, false);
}

__device__ __forceinline__ float sigm(float x) { return 1.f / (1.f + expf(-x)); }

// Async 16-byte global -> LDS copy (per active lane). lds = DS byte address
// (= low 32 bits of the flat shared-memory pointer on gfx1250).
__device__ __forceinline__ void async_cp16(uint32_t lds, const float* g) {
  asm volatile("global_load_async_to_lds_b128 %0, %1, off"
               :
               : "v"(lds), "v"(g)
               : "memory");
}
__device__ __forceinline__ void wait_async0() {
  asm volatile("s_wait_asynccnt 0" ::: "memory");
}
__device__ __forceinline__ uint32_t lds_addr(const void* p) {
  return (uint32_t)(uintptr_t)p;   // flat LDS address: low 32 bits = DS offset
}

// ---------------- K1: three conv branches (k=3,5,1 over T), raw pre-BN ------
__global__ void k_conv(const float* __restrict__ X, const float* __restrict__ w2,
                       const float* __restrict__ b2, const float* __restrict__ w3,
                       const float* __restrict__ b3, const float* __restrict__ w4,
                       const float* __restrict__ b4, float* __restrict__ y) {
  int idx = blockIdx.x * blockDim.x + threadIdx.x;
  if (idx >= BB * NN * TT * DD) return;
  int c = idx & 63;
  int tf = idx >> 6;
  int t = tf & 15;
  int bn = tf >> 4;
  const float* xr = X + (size_t)bn * TT * 3;   // [16][3] slab for this (b,n)
  float acc;
  if (c < 32) {                      // conv2: k=3, pad=1 -> 32 channels
    const float* w = w2 + c * 9;     // [i][j] = i*3+j
    acc = b2[c];
#pragma unroll
    for (int j = 0; j < 3; ++j) {
      int tau = t + j - 1;
      if (tau < 0 || tau >= TT) continue;
#pragma unroll
      for (int i = 0; i < 3; ++i) acc += xr[tau * 3 + i] * w[i * 3 + j];
    }
  } else if (c < 48) {               // conv3: k=5, pad=2 -> 16 channels
    int o = c - 32;
    const float* w = w3 + o * 15;    // [i][j] = i*5+j
    acc = b3[o];
#pragma unroll
    for (int j = 0; j < 5; ++j) {
      int tau = t + j - 2;
      if (tau < 0 || tau >= TT) continue;
#pragma unroll
      for (int i = 0; i < 3; ++i) acc += xr[tau * 3 + i] * w[i * 5 + j];
    }
  } else {                           // conv4: k=1 -> 16 channels
    int o = c - 48;
    const float* w = w4 + o * 3;
    acc = b4[o];
#pragma unroll
    for (int i = 0; i < 3; ++i) acc += xr[t * 3 + i] * w[i];
  }
  y[idx] = acc;
}

// ---------------- K2: per-channel BN stats -> affine (a, c) ----------------
__global__ void k_chanbn(const float* __restrict__ y, const float* g2,
                         const float* be2, const float* g3, const float* be3,
                         const float* g4, const float* be4,
                         float* __restrict__ chanAC) {
  int c = blockIdx.x;   // 64 blocks
  int tid = threadIdx.x;
  float s = 0.f, ss = 0.f;
  for (int k = tid; k < BB * NN * TT; k += 256) {
    float v = y[(size_t)k * 64 + c];
    s += v;
    ss += v * v;
  }
  __shared__ float rs[256], rq[256];
  rs[tid] = s; rq[tid] = ss;
  __syncthreads();
  for (int o = 128; o > 0; o >>= 1) {
    if (tid < o) { rs[tid] += rs[tid + o]; rq[tid] += rq[tid + o]; }
    __syncthreads();
  }
  if (tid == 0) {
    const float M = (float)(BB * NN * TT);
    float mu = rs[0] / M;
    float var = rq[0] / M - mu * mu;
    float g, b;
    if (c < 32)       { g = g2[c];      b = be2[c]; }
    else if (c < 48)  { g = g3[c - 32]; b = be3[c - 32]; }
    else              { g = g4[c - 48]; b = be4[c - 48]; }
    float a = g * rsqrtf(var + 1e-3f);
    chanAC[c] = a;
    chanAC[64 + c] = b - mu * a;
  }
}

// ---------------- K3: BN+ReLU in place, GAP over (N,T) per (b,c) -----------
__global__ void k_bnrelu_gap(float* __restrict__ y,
                             const float* __restrict__ chanAC,
                             float* __restrict__ gap) {
  int b = blockIdx.x >> 6;
  int c = blockIdx.x & 63;
  int tid = threadIdx.x;
  float a = chanAC[c], cc = chanAC[64 + c];
  float s = 0.f;
  size_t base = (size_t)b * NN * TT * 64 + c;
  for (int k = tid; k < NN * TT; k += 256) {
    size_t id = base + (size_t)k * 64;
    float v = fmaxf(a * y[id] + cc, 0.f);
    y[id] = v;
    s += v;
  }
  __shared__ float rs[256];
  rs[tid] = s;
  __syncthreads();
  for (int o = 128; o > 0; o >>= 1) {
    if (tid < o) rs[tid] += rs[tid + o];
    __syncthreads();
  }
  if (tid == 0) gap[b * 64 + c] = rs[0] / (float)(NN * TT);
}

// ---------------- K4: ECA channel attention --------------------------------
__global__ void k_eca(const float* __restrict__ gap, const float* __restrict__ ew,
                      float* __restrict__ att) {
  int idx = blockIdx.x * blockDim.x + threadIdx.x;
  if (idx >= BB * 64) return;
  int b = idx >> 6, d = idx & 63;
  float l = (d > 0) ? gap[b * 64 + d - 1] : 0.f;
  float m = gap[b * 64 + d];
  float r = (d < 63) ? gap[b * 64 + d + 1] : 0.f;
  att[idx] = sigm(ew[0] * l + ew[1] * m + ew[2] * r);
}

// ---------------- K5: apply gate + Linear T->12, write t [B,N,12,64] -------
__global__ void k_tlin(const float* __restrict__ cat, const float* __restrict__ att,
                       const float* __restrict__ tw, const float* __restrict__ tb,
                       float* __restrict__ tbuf) {
  int idx = blockIdx.x * blockDim.x + threadIdx.x;
  if (idx >= BB * NN * TO * DD) return;
  int d = idx & 63;
  int s = (idx >> 6) % TO;
  int bn = idx / (TO * DD);
  int b = bn >> 9;
  float av = att[b * 64 + d];
  float acc = tb[s];
  const float* cr = cat + (size_t)bn * TT * 64 + d;
#pragma unroll
  for (int t = 0; t < TT; ++t) acc += fmaxf(cr[t * 64] * av, 0.f) * tw[s * TT + t];
  tbuf[idx] = acc;
}

// ---------------- K6/K7/K8: spatial attention inputs -----------------------
__global__ void k_o1(const float* __restrict__ tbuf, const float* __restrict__ w1,
                     float* __restrict__ o1) {
  int idx = blockIdx.x * blockDim.x + threadIdx.x;
  if (idx >= BB * NN * DD) return;
  int d = idx & 63;
  int bn = idx >> 6;
  const float* tr = tbuf + (size_t)bn * TO * 64 + d;
  float acc = 0.f;
#pragma unroll
  for (int s = 0; s < TO; ++s) acc += tr[s * 64] * w1[s];
  o1[idx] = acc;
}

__global__ void k_o2(const float* __restrict__ o1, const float* __restrict__ w2,
                     float* __restrict__ o2) {
  int idx = blockIdx.x * blockDim.x + threadIdx.x;
  if (idx >= BB * NN * TO) return;
  int s = idx % TO;
  int bn = idx / TO;
  const float* orow = o1 + (size_t)bn * 64;
  float acc = 0.f;
#pragma unroll
  for (int d = 0; d < 64; ++d) acc += orow[d] * w2[d * TO + s];
  o2[idx] = acc;
}

__global__ void k_o3(const float* __restrict__ tbuf, const float* __restrict__ w3,
                     float* __restrict__ o3) {
  int idx = blockIdx.x * blockDim.x + threadIdx.x;
  if (idx >= BB * TO * NN) return;
  int n = idx & 511;
  int rest = idx >> 9;
  int s = rest % TO;
  int b = rest / TO;
  const float* tr = tbuf + ((size_t)b * ROWS_PER_B + n * TO + s) * 64;
  float acc = 0.f;
#pragma unroll
  for (int d = 0; d < 64; ++d) acc += tr[d] * w3[d];
  o3[(b * TO + s) * NN + n] = acc;
}

// ---------------- K9: WMMA  S = sigmoid(o2 @ o3 + e1)  [512x12]x[12x512] ---
__global__ void __launch_bounds__(32) k_scores(const float* __restrict__ o2,
                                               const float* __restrict__ o3,
                                               const float* __restrict__ e1,
                                               float* __restrict__ S) {
  int lane = threadIdx.x;
  int nb = blockIdx.x, mb = blockIdx.y, b = blockIdx.z;
  int r = lane & 15, h = lane >> 4;
  const float* A = o2 + (size_t)b * NN * TO;
  const float* Bm = o3 + (size_t)b * TO * NN;
  v8f acc = {0.f, 0.f, 0.f, 0.f, 0.f, 0.f, 0.f, 0.f};
#pragma unroll
  for (int kk = 0; kk < 3; ++kk) {          // K = 12 = 3 * 4
    int k0 = kk * 4 + 2 * h;
    v2f a, bf;
    a.x = A[(mb * 16 + r) * TO + k0];
    a.y = A[(mb * 16 + r) * TO + k0 + 1];
    bf.x = Bm[k0 * NN + nb * 16 + r];
    bf.y = Bm[(k0 + 1) * NN + nb * 16 + r];
    acc = wmma4(a, bf, acc);
  }
  size_t base = (size_t)b * NN * NN;
#pragma unroll
  for (int v = 0; v < 8; ++v) {
    int m = mb * 16 + v + 8 * h;
    int n = nb * 16 + r;
    S[base + (size_t)m * NN + n] = sigm(acc[v] + e1[m * NN + n]);
  }
}

// ---------------- K10: WMMA  E = e2 @ S  (512^3 per batch) -----------------
// 64x64 tile per 128-thread block (4 waves, each owns a 16-row strip).
// K staged in 32-deep chunks through double-buffered LDS via async copies.
__global__ void __launch_bounds__(128) k_egemm(const float* __restrict__ e2,
                                               const float* __restrict__ S,
                                               float* __restrict__ E) {
  __shared__ float As[2][64 * 32];   // A chunk: 64 rows x 32 k
  __shared__ float Bs[2][32 * 64];   // B chunk: 32 k x 64 cols
  int tid = threadIdx.x;
  int lane = tid & 31, w = tid >> 5;
  int r = lane & 15, h = lane >> 4;
  int nb = blockIdx.x, mb = blockIdx.y, b = blockIdx.z;
  const float* Sb = S + (size_t)b * NN * NN;
  const float* A0 = e2 + (size_t)(mb * 64) * NN;   // row-major [64][512]
  const float* B0 = Sb + nb * 64;                  // rows k, stride 512

  v8f acc[4];
#pragma unroll
  for (int nt = 0; nt < 4; ++nt)
    acc[nt] = (v8f){0.f, 0.f, 0.f, 0.f, 0.f, 0.f, 0.f, 0.f};

  // stage K-chunk kc (32 wide) into buffer `buf`: 8 async b128 per thread
  auto stage = [&](int buf, int kc) {
#pragma unroll
    for (int i = 0; i < 4; ++i) {               // A: 512 float4 total
      int f = i * 128 + tid;
      int row = f >> 3, q = f & 7;
      async_cp16(lds_addr(&As[buf][row * 32 + q * 4]),
                 A0 + (size_t)row * NN + kc + q * 4);
    }
#pragma unroll
    for (int i = 0; i < 4; ++i) {               // B: 512 float4 total
      int f = i * 128 + tid;
      int row = f >> 4, q = f & 15;
      async_cp16(lds_addr(&Bs[buf][row * 64 + q * 4]),
                 B0 + (size_t)(kc + row) * NN + q * 4);
    }
  };

  stage(0, 0);
  wait_async0();
  __syncthreads();

  int buf = 0;
  for (int c = 0; c < 16; ++c) {                // 16 chunks x 32 = K 512
    if (c + 1 < 16) stage(buf ^ 1, (c + 1) * 32);
    const float* AsB = As[buf];
    const float* BsB = Bs[buf];
#pragma unroll
    for (int kk = 0; kk < 8; ++kk) {
      int k0 = kk * 4 + 2 * h;
      v2f a = *(const v2f*)(AsB + (w * 16 + r) * 32 + k0);
#pragma unroll
      for (int nt = 0; nt < 4; ++nt) {
        v2f bf;
        bf.x = BsB[k0 * 64 + nt * 16 + r];
        bf.y = BsB[(k0 + 1) * 64 + nt * 16 + r];
        acc[nt] = wmma4(a, bf, acc[nt]);
      }
    }
    wait_async0();        // chunk c+1 landed
    __syncthreads();      // all waves done reading buf / writing buf^1
    buf ^= 1;
  }

  size_t base = (size_t)b * NN * NN;
#pragma unroll
  for (int nt = 0; nt < 4; ++nt)
#pragma unroll
    for (int v = 0; v < 8; ++v) {
      int m = mb * 64 + w * 16 + v + 8 * h;
      int n = nb * 64 + nt * 16 + r;
      E[base + (size_t)m * NN + n] = acc[nt][v];
    }
}

// ---------------- K11: column softmax, keep only diagonal ------------------
__global__ void k_diagsm(const float* __restrict__ E, float* __restrict__ dsat) {
  int n = blockIdx.x, b = blockIdx.y, tid = threadIdx.x;
  const float* col = E + (size_t)b * NN * NN + n;
  float v0 = col[(size_t)tid * NN];
  float v1 = col[(size_t)(tid + 256) * NN];
  __shared__ float red[256];
  red[tid] = fmaxf(v0, v1);
  __syncthreads();
  for (int o = 128; o > 0; o >>= 1) {
    if (tid < o) red[tid] = fmaxf(red[tid], red[tid + o]);
    __syncthreads();
  }
  float mx = red[0];
  __syncthreads();
  red[tid] = expf(v0 - mx) + expf(v1 - mx);
  __syncthreads();
  for (int o = 128; o > 0; o >>= 1) {
    if (tid < o) red[tid] += red[tid + o];
    __syncthreads();
  }
  if (tid == 0) dsat[b * NN + n] = expf(col[(size_t)n * NN] - mx) / red[0];
}

// ---------------- K12: adjacency diagonals ---------------------------------
__global__ void k_adjd(const float* __restrict__ adj, float* __restrict__ adjd) {
  int idx = blockIdx.x * blockDim.x + threadIdx.x;
  if (idx >= 3 * NN) return;
  int g = idx / NN, n = idx % NN;
  adjd[idx] = adj[(size_t)g * NN * NN + (size_t)n * NN + n];
}

// ---------------- K13: fused WMMA chain: 3 graph blocks + Theta + gate -----
// One wave per 16-row tile of the [B, N*12, 64] activation tensor.
__global__ void __launch_bounds__(32) k_graph(
    const float* __restrict__ tbuf, const float* __restrict__ adjd,
    const float* __restrict__ dsat, const float* __restrict__ gbw,
    const float* __restrict__ gbb, const float* __restrict__ Theta,
    const float* __restrict__ l2w, const float* __restrict__ l2b,
    float* __restrict__ t3) {
  __shared__ float xs[16 * 64];    // original t tile (kept for gate + blend)
  __shared__ float hs[16 * 64];    // running "his"
  __shared__ float og[16 * 192];   // concat(og1, og2, og3)
  __shared__ float t2s[16 * 64];   // relu(outg @ Theta1)
  __shared__ float dr[3][16];      // per-row diag scales

  int lane = threadIdx.x;
  int tile = blockIdx.x;           // 384 tiles per batch
  int b = blockIdx.y;
  int r = lane & 15, h = lane >> 4;
  size_t gbase = ((size_t)b * ROWS_PER_B + (size_t)tile * 16) * 64;

  // L2-hot weights: hint the prefetcher (global_prefetch_b8)
  __builtin_prefetch(gbw + lane * 64, 0, 1);
  __builtin_prefetch(Theta + lane * 64, 0, 1);
  __builtin_prefetch(l2w + lane * 64, 0, 1);

  // contiguous 1024-float tile load
#pragma unroll
  for (int i = 0; i < 32; ++i) xs[i * 32 + lane] = tbuf[gbase + i * 32 + lane];
  if (lane < 16) {
    int gr = tile * 16 + lane;
    int node = gr / TO;
    float ds = dsat[b * NN + node];
    dr[0][lane] = adjd[node] * ds;   // d0 = diag(adj0) * diag(sat)
    dr[1][lane] = adjd[NN + node];
    dr[2][lane] = adjd[2 * NN + node];
  }
  __syncthreads();

  // ---- three graph blocks: G = relu((d .* src) @ W^T + bias) [16x128] ----
  for (int g = 0; g < 3; ++g) {
    const float* W = gbw + g * 8192;      // [128][64]
    const float* bias = gbb + g * 128;
    const float* src = (g == 0) ? xs : hs;
    float dscale = dr[g][r];
    v8f acc[8];
#pragma unroll
    for (int nt = 0; nt < 8; ++nt) acc[nt] = (v8f){0.f,0.f,0.f,0.f,0.f,0.f,0.f,0.f};
    for (int kk = 0; kk < 16; ++kk) {     // K = 64
      int k0 = kk * 4 + 2 * h;
      v2f a;
      a.x = src[r * 64 + k0] * dscale;
      a.y = src[r * 64 + k0 + 1] * dscale;
#pragma unroll
      for (int nt = 0; nt < 8; ++nt) {    // N = 128
        v2f bf;
        bf.x = W[(nt * 16 + r) * 64 + k0];
        bf.y = W[(nt * 16 + r) * 64 + k0 + 1];
        acc[nt] = wmma4(a, bf, acc[nt]);
      }
    }
    __syncthreads();
#pragma unroll
    for (int nt = 0; nt < 8; ++nt) {
      int o = nt * 16 + r;
      float bb = bias[o];
#pragma unroll
      for (int v = 0; v < 8; ++v) {
        int m = v + 8 * h;
        float gval = fmaxf(acc[nt][v] + bb, 0.f);
        if (o < 64) hs[m * 64 + o] = src[m * 64 + o] - gval;   // his update
        else og[m * 192 + g * 64 + (o - 64)] = gval;           // concat slot
      }
    }
    __syncthreads();
  }

  // ---- t2 = relu(outg @ Theta1)  [16x192]x[192x64] ----
  {
    v8f acc[4];
#pragma unroll
    for (int nt = 0; nt < 4; ++nt) acc[nt] = (v8f){0.f,0.f,0.f,0.f,0.f,0.f,0.f,0.f};
    for (int kk = 0; kk < 48; ++kk) {     // K = 192
      int k0 = kk * 4 + 2 * h;
      v2f a;
      a.x = og[r * 192 + k0];
      a.y = og[r * 192 + k0 + 1];
#pragma unroll
      for (int nt = 0; nt < 4; ++nt) {
        v2f bf;
        bf.x = Theta[k0 * 64 + nt * 16 + r];
        bf.y = Theta[(k0 + 1) * 64 + nt * 16 + r];
        acc[nt] = wmma4(a, bf, acc[nt]);
      }
    }
#pragma unroll
    for (int nt = 0; nt < 4; ++nt)
#pragma unroll
      for (int v = 0; v < 8; ++v) {
        int m = v + 8 * h, c = nt * 16 + r;
        t2s[m * 64 + c] = fmaxf(acc[nt][v], 0.f);
      }
  }
  __syncthreads();

  // ---- rgate = sigmoid([t2|x] @ l2w^T + l2b); t3 = (1-r)*t2 + r*x ----
  {
    v8f acc[4];
#pragma unroll
    for (int nt = 0; nt < 4; ++nt) acc[nt] = (v8f){0.f,0.f,0.f,0.f,0.f,0.f,0.f,0.f};
    for (int kk = 0; kk < 32; ++kk) {     // K = 128
      int k0 = kk * 4 + 2 * h;
      const float* srcA = (k0 < 64) ? t2s : xs;
      int ko = (k0 < 64) ? k0 : k0 - 64;
      v2f a;
      a.x = srcA[r * 64 + ko];
      a.y = srcA[r * 64 + ko + 1];
#pragma unroll
      for (int nt = 0; nt < 4; ++nt) {
        v2f bf;
        bf.x = l2w[(nt * 16 + r) * 128 + k0];
        bf.y = l2w[(nt * 16 + r) * 128 + k0 + 1];
        acc[nt] = wmma4(a, bf, acc[nt]);
      }
    }
#pragma unroll
    for (int nt = 0; nt < 4; ++nt)
#pragma unroll
      for (int v = 0; v < 8; ++v) {
        int m = v + 8 * h, c = nt * 16 + r;
        float rg = sigm(acc[nt][v] + l2b[c]);
        float t2v = t2s[m * 64 + c];
        float tv = xs[m * 64 + c];
        t3[gbase + m * 64 + c] = (1.f - rg) * t2v + rg * tv;
      }
  }
}

// ---------------- K14: per-node BN stats -> affine -------------------------
__global__ void k_bnf(const float* __restrict__ t3, const float* __restrict__ g,
                      const float* __restrict__ be, float* __restrict__ bnac) {
  int n = blockIdx.x, tid = threadIdx.x;
  float s = 0.f, ss = 0.f;
  for (int k = tid; k < BB * TO * DD; k += 256) {   // 49152 per node
    int bb = k / (TO * DD);
    int rem = k % (TO * DD);
    float v = t3[(size_t)bb * ROWS_PER_B * 64 + (size_t)n * TO * 64 + rem];
    s += v;
    ss += v * v;
  }
  __shared__ float rs[256], rq[256];
  rs[tid] = s; rq[tid] = ss;
  __syncthreads();
  for (int o = 128; o > 0; o >>= 1) {
    if (tid < o) { rs[tid] += rs[tid + o]; rq[tid] += rq[tid + o]; }
    __syncthreads();
  }
  if (tid == 0) {
    const float M = (float)(BB * TO * DD);
    float mu = rs[0] / M;
    float var = rq[0] / M - mu * mu;
    float a = g[n] * rsqrtf(var + 1e-5f);
    bnac[n] = a;
    bnac[NN + n] = be[n] - mu * a;
  }
}

// ---------------- K15: final normalize -> d_out ----------------------------
__global__ void k_out(const float* __restrict__ t3, const float* __restrict__ bnac,
                      float* __restrict__ out) {
  int idx = blockIdx.x * blockDim.x + threadIdx.x;
  if (idx >= BB * NN * TO * DD) return;
  int n = (idx / (TO * DD)) % NN;
  out[idx] = bnac[n] * t3[idx] + bnac[NN + n];
}

// ---------------------------------------------------------------------------
extern "C" void kernel_launch(void* const* d_in, const int* in_sizes, int n_in,
                              void* d_out, int out_size, void* d_ws, size_t ws_size,
                              hipStream_t stream) {
  (void)in_sizes; (void)n_in; (void)out_size; (void)ws_size;
  const float* X    = (const float*)d_in[0];
  const float* adj  = (const float*)d_in[1];
  const float* c2w  = (const float*)d_in[2];
  const float* c2b  = (const float*)d_in[3];
  const float* bn2g = (const float*)d_in[4];
  const float* bn2b = (const float*)d_in[5];
  const float* c3w  = (const float*)d_in[6];
  const float* c3b  = (const float*)d_in[7];
  const float* bn3g = (const float*)d_in[8];
  const float* bn3b = (const float*)d_in[9];
  const float* c4w  = (const float*)d_in[10];
  const float* c4b  = (const float*)d_in[11];
  const float* bn4g = (const float*)d_in[12];
  const float* bn4b = (const float*)d_in[13];
  const float* ecaw = (const float*)d_in[14];
  const float* tlw  = (const float*)d_in[15];
  const float* tlb  = (const float*)d_in[16];
  const float* w1   = (const float*)d_in[17];
  const float* w2   = (const float*)d_in[18];
  const float* w3   = (const float*)d_in[19];
  const float* e1   = (const float*)d_in[20];
  const float* e2   = (const float*)d_in[21];
  const float* gbw  = (const float*)d_in[22];
  const float* gbb  = (const float*)d_in[23];
  const float* Th   = (const float*)d_in[24];
  const float* l2w  = (const float*)d_in[25];
  const float* l2b  = (const float*)d_in[26];
  const float* bnfg = (const float*)d_in[27];
  const float* bnfb = (const float*)d_in[28];

  float* ws = (float*)d_ws;
  // region A (33,554,432 fl): y/cat in phase 1; reused for S + E in phase 2
  float* ybuf = ws;
  float* Sbuf = ws;
  float* Ebuf = ws + 16777216;
  float* tbuf = ws + 33554432;   // 25,165,824 fl  [B,N,12,64]
  float* t3   = ws + 58720256;   // 25,165,824 fl
  float* o1   = ws + 83886080;   //  2,097,152 fl
  float* o2   = ws + 85983232;   //    393,216 fl
  float* o3   = ws + 86376448;   //    393,216 fl
  float* gap  = ws + 86769664;   //      4,096 fl
  float* att  = ws + 86773760;   //      4,096 fl
  float* chan = ws + 86777856;   //        128 fl
  float* dsat = ws + 86777984;   //     32,768 fl
  float* adjd = ws + 86810752;   //      1,536 fl
  float* bnac = ws + 86812288;   //      1,024 fl

  // phase 1: TimeBlock
  k_conv<<<(BB * NN * TT * DD + 255) / 256, 256, 0, stream>>>(
      X, c2w, c2b, c3w, c3b, c4w, c4b, ybuf);
  k_chanbn<<<64, 256, 0, stream>>>(ybuf, bn2g, bn2b, bn3g, bn3b, bn4g, bn4b, chan);
  k_bnrelu_gap<<<BB * 64, 256, 0, stream>>>(ybuf, chan, gap);
  k_eca<<<(BB * 64 + 255) / 256, 256, 0, stream>>>(gap, ecaw, att);
  k_tlin<<<(BB * NN * TO * DD + 255) / 256, 256, 0, stream>>>(ybuf, att, tlw, tlb,
                                                              tbuf);
  // phase 2: spatial attention (ybuf dead; region reused for S/E)
  k_o1<<<(BB * NN * DD + 255) / 256, 256, 0, stream>>>(tbuf, w1, o1);
  k_o2<<<(BB * NN * TO + 255) / 256, 256, 0, stream>>>(o1, w2, o2);
  k_o3<<<(BB * TO * NN + 255) / 256, 256, 0, stream>>>(tbuf, w3, o3);
  k_scores<<<dim3(NN / 16, NN / 16, BB), 32, 0, stream>>>(o2, o3, e1, Sbuf);
  k_egemm<<<dim3(NN / 64, NN / 64, BB), 128, 0, stream>>>(e2, Sbuf, Ebuf);
  k_diagsm<<<dim3(NN, BB), 256, 0, stream>>>(Ebuf, dsat);
  k_adjd<<<(3 * NN + 255) / 256, 256, 0, stream>>>(adj, adjd);
  // phase 3: fused graph chain (3 gblocks + Theta + gated fusion)
  k_graph<<<dim3(ROWS_PER_B / 16, BB), 32, 0, stream>>>(tbuf, adjd, dsat, gbw, gbb,
                                                        Th, l2w, l2b, t3);
  // phase 4: per-node BatchNorm
  k_bnf<<<NN, 256, 0, stream>>>(t3, bnfg, bnfb, bnac);
  k_out<<<(BB * NN * TO * DD + 255) / 256, 256, 0, stream>>>(t3, bnac,
                                                             (float*)d_out);
}